// Decomp_Decoders_64312840290453
// MI455X (gfx1250) — compile-verified
//
#include <hip/hip_runtime.h>
#include <hip/hip_bf16.h>

typedef __attribute__((ext_vector_type(2))) float v2f;
typedef __attribute__((ext_vector_type(8))) float v8f;

#define NFREQ  6
#define CCH    16
#define STRIDE 74      // LDS row stride (floats): even (8B-aligned b64 loads), odd/64-coprime word stride /2 for bank spread
#define WAVES  4
#define BLOCK  128     // 4 wave32 per block; each wave owns 32 points (two 16-row WMMA M-tiles)

struct Params {
  const float* p;
  const float* bound;
  const float* pl[12];   // geo_xy0,geo_xz0,geo_yz0,geo_xy1,geo_xz1,geo_yz1, c_xy0,...,c_yz1
  const float* W0; const float* b0; const float* W1; const float* b1; const float* Wo; const float* bo;
  const float* cW0; const float* cb0; const float* cW1; const float* cb1; const float* cWo; const float* cbo;
  float* out;
  int n;
};

__device__ __forceinline__ v8f wmma_f32(v2f a, v2f b, v8f c) {
  // D = A(16x4) * B(4x16) + C(16x16), all f32, full precision accumulate
  return __builtin_amdgcn_wmma_f32_16x16x4_f32(false, a, false, b, (short)0, c, false, false);
}

__device__ __forceinline__ void lds_fence() {
  // Same-wave cross-lane LDS RAW/WAR fence: waits all outstanding DS ops and
  // blocks compiler reordering. Wave-private LDS regions -> no block barrier needed.
  asm volatile("s_wait_dscnt 0" ::: "memory");
}

// Bilinear grid_sample (border clamp, align_corners=true) for one (1,16,R,R) plane.
__device__ __forceinline__ void sample_plane(const float* __restrict__ pl, int R,
                                             float gx, float gy, float* acc) {
  const float fm1 = (float)(R - 1);
  float ix = fminf(fmaxf((gx + 1.0f) * 0.5f * fm1, 0.0f), fm1);
  float iy = fminf(fmaxf((gy + 1.0f) * 0.5f * fm1, 0.0f), fm1);
  float fx0 = floorf(ix), fy0 = floorf(iy);
  int x0 = (int)fx0, y0 = (int)fy0;
  int x1 = min(x0 + 1, R - 1);
  int y1 = min(y0 + 1, R - 1);
  float wx = ix - fx0, wy = iy - fy0;
  float w00 = (1.0f - wx) * (1.0f - wy), w01 = wx * (1.0f - wy);
  float w10 = (1.0f - wx) * wy,          w11 = wx * wy;
  const int HW = R * R;
  int i00 = y0 * R + x0, i01 = y0 * R + x1, i10 = y1 * R + x0, i11 = y1 * R + x1;
  #pragma unroll
  for (int c = 0; c < CCH; ++c) {
    const float* b = pl + c * HW;
    acc[c] = fmaf(w00, b[i00], fmaf(w01, b[i01], fmaf(w10, b[i10], fmaf(w11, b[i11], acc[c]))));
  }
}

// Two-layer MLP (71->32 relu, 32->32 relu) for this wave's 32 points via f32 WMMA.
// Inputs: wave rows cols 0..71 (col 71 zero-padded). Output h2 relu'd into cols 40..71.
__device__ __forceinline__ void run_mlp(float* wrow, const v2f* sw0, const v2f* sw1,
                                        const float* __restrict__ bias0,
                                        const float* __restrict__ bias1, int lane) {
  const int half = lane >> 4;   // K sub-tile select in A/B fragments
  const int ncol = lane & 15;   // N column within a 16-wide tile / M row for A
  const float b00 = bias0[ncol], b01 = bias0[ncol + 16];
  const float b10 = bias1[ncol], b11 = bias1[ncol + 16];
  #pragma unroll
  for (int tm = 0; tm < 2; ++tm) {          // two M=16 point tiles per wave
    float* tb = wrow + tm * 16 * STRIDE;
    // ---- layer 0: K = 72 (18 steps of 4), N tiles {0..15},{16..31} ----
    v8f c0, c1;
    #pragma unroll
    for (int r = 0; r < 8; ++r) { c0[r] = b00; c1[r] = b01; }
    const float* arow = tb + ncol * STRIDE + 2 * half;   // A: M=ncol, K=k0+2*half
    #pragma unroll
    for (int k = 0; k < 18; ++k) {
      v2f a  = *(const v2f*)(arow + 4 * k);
      v2f bA = sw0[(2 * k + half) * 32 + ncol];          // B: K-pair rows, N=ncol(+16)
      v2f bB = sw0[(2 * k + half) * 32 + ncol + 16];
      c0 = wmma_f32(a, bA, c0);
      c1 = wmma_f32(a, bB, c1);
    }
    #pragma unroll
    for (int r = 0; r < 8; ++r) {           // D row M = r + 8*half; restage h1 at cols 40..71
      tb[(r + 8 * half) * STRIDE + 40 + ncol]      = fmaxf(c0[r], 0.0f);
      tb[(r + 8 * half) * STRIDE + 40 + ncol + 16] = fmaxf(c1[r], 0.0f);
    }
    lds_fence();
    // ---- layer 1: K = 32 (8 steps of 4) reading cols 40..71 ----
    v8f d0, d1;
    #pragma unroll
    for (int r = 0; r < 8; ++r) { d0[r] = b10; d1[r] = b11; }
    const float* arow2 = tb + ncol * STRIDE + 40 + 2 * half;
    #pragma unroll
    for (int k = 0; k < 8; ++k) {
      v2f a  = *(const v2f*)(arow2 + 4 * k);
      v2f bA = sw1[(2 * k + half) * 32 + ncol];
      v2f bB = sw1[(2 * k + half) * 32 + ncol + 16];
      d0 = wmma_f32(a, bA, d0);
      d1 = wmma_f32(a, bB, d1);
    }
    #pragma unroll
    for (int r = 0; r < 8; ++r) {           // h2 (relu) overwrites cols 40..71
      tb[(r + 8 * half) * STRIDE + 40 + ncol]      = fmaxf(d0[r], 0.0f);
      tb[(r + 8 * half) * STRIDE + 40 + ncol + 16] = fmaxf(d1[r], 0.0f);
    }
  }
  lds_fence();
}

__global__ void __launch_bounds__(BLOCK) decomp_decoders_kernel(Params prm) {
  __shared__ __align__(16) float sH[WAVES * 32 * STRIDE];  // 37,888 B feature/hidden staging
  __shared__ v2f sW0g[36 * 32];                            // W0^T geo, K-pair packed (9,216 B)
  __shared__ v2f sW1g[16 * 32];                            // W1^T geo (4,096 B)
  __shared__ v2f sW0c[36 * 32];                            // color
  __shared__ v2f sW1c[16 * 32];                            // total LDS = 64,512 B

  const int tid = threadIdx.x;

  // ---- stage transposed, K-pair-packed weights (K padded 71 -> 72 with zeros) ----
  for (int i = tid; i < 36 * 32; i += BLOCK) {
    int n = i & 31, k = (i >> 5) * 2;
    v2f t;
    t.x = (k     < 71) ? prm.W0[n * 71 + k]     : 0.0f;
    t.y = (k + 1 < 71) ? prm.W0[n * 71 + k + 1] : 0.0f;
    sW0g[i] = t;
    t.x = (k     < 71) ? prm.cW0[n * 71 + k]     : 0.0f;
    t.y = (k + 1 < 71) ? prm.cW0[n * 71 + k + 1] : 0.0f;
    sW0c[i] = t;
  }
  for (int i = tid; i < 16 * 32; i += BLOCK) {
    int n = i & 31, k = (i >> 5) * 2;
    v2f t;
    t.x = prm.W1[n * 32 + k]; t.y = prm.W1[n * 32 + k + 1];
    sW1g[i] = t;
    t.x = prm.cW1[n * 32 + k]; t.y = prm.cW1[n * 32 + k + 1];
    sW1c[i] = t;
  }
  __syncthreads();

  const int lane = tid & 31;
  const int wave = tid >> 5;
  float* wrow  = sH + wave * 32 * STRIDE;   // wave-private 32-row region
  float* myrow = wrow + lane * STRIDE;      // this point's feature row

  const int gp  = blockIdx.x * BLOCK + tid;
  const int gpc = min(gp, prm.n - 1);       // clamp for tail (uniform WMMA, guarded stores)

  // ---- normalize + positional embedding (cols 0..38), zero pad col 71 ----
  const float bx0 = prm.bound[0], bx1 = prm.bound[1];
  const float by0 = prm.bound[2], by1 = prm.bound[3];
  const float bz0 = prm.bound[4], bz1 = prm.bound[5];
  const float px = (prm.p[gpc * 3 + 0] - bx0) / (bx1 - bx0);
  const float py = (prm.p[gpc * 3 + 1] - by0) / (by1 - by0);
  const float pz = (prm.p[gpc * 3 + 2] - bz0) / (bz1 - bz0);
  myrow[0] = px; myrow[1] = py; myrow[2] = pz;
  #pragma unroll
  for (int j = 0; j < NFREQ; ++j) {
    const float f = 3.14159265358979323846f * (float)(1 << j);
    float s, c;
    sincosf(px * f, &s, &c); myrow[3 + j * 3 + 0] = s; myrow[21 + j * 3 + 0] = c;
    sincosf(py * f, &s, &c); myrow[3 + j * 3 + 1] = s; myrow[21 + j * 3 + 1] = c;
    sincosf(pz * f, &s, &c); myrow[3 + j * 3 + 2] = s; myrow[21 + j * 3 + 2] = c;
  }
  myrow[71] = 0.0f;

  float acc[CCH];

  // ================= geometry branch =================
  #pragma unroll
  for (int c = 0; c < CCH; ++c) acc[c] = 0.0f;
  sample_plane(prm.pl[0], 512,  px, py, acc);
  sample_plane(prm.pl[1], 512,  px, pz, acc);
  sample_plane(prm.pl[2], 512,  py, pz, acc);
  #pragma unroll
  for (int c = 0; c < CCH; ++c) myrow[39 + c] = acc[c];
  #pragma unroll
  for (int c = 0; c < CCH; ++c) acc[c] = 0.0f;
  sample_plane(prm.pl[3], 1024, px, py, acc);
  sample_plane(prm.pl[4], 1024, px, pz, acc);
  sample_plane(prm.pl[5], 1024, py, pz, acc);
  #pragma unroll
  for (int c = 0; c < CCH; ++c) myrow[55 + c] = acc[c];

  lds_fence();
  run_mlp(wrow, sW0g, sW1g, prm.b0, prm.b1, lane);

  {  // sdf head: tanh(h2 . Wo + bo)
    float a = prm.bo[0];
    #pragma unroll
    for (int i = 0; i < 32; ++i) a = fmaf(myrow[40 + i], prm.Wo[i], a);
    if (gp < prm.n) prm.out[gp * 4 + 3] = tanhf(a);
  }

  // ================= color branch =================
  #pragma unroll
  for (int c = 0; c < CCH; ++c) acc[c] = 0.0f;
  sample_plane(prm.pl[6], 512,  px, py, acc);
  sample_plane(prm.pl[7], 512,  px, pz, acc);
  sample_plane(prm.pl[8], 512,  py, pz, acc);
  #pragma unroll
  for (int c = 0; c < CCH; ++c) myrow[39 + c] = acc[c];
  #pragma unroll
  for (int c = 0; c < CCH; ++c) acc[c] = 0.0f;
  sample_plane(prm.pl[9],  1024, px, py, acc);
  sample_plane(prm.pl[10], 1024, px, pz, acc);
  sample_plane(prm.pl[11], 1024, py, pz, acc);
  #pragma unroll
  for (int c = 0; c < CCH; ++c) myrow[55 + c] = acc[c];
  myrow[71] = 0.0f;   // was h2[31] from geo pass

  lds_fence();
  run_mlp(wrow, sW0c, sW1c, prm.cb0, prm.cb1, lane);

  #pragma unroll
  for (int r = 0; r < 3; ++r) {   // rgb head: h2 . cWo^T + cbo (no activation)
    float a = prm.cbo[r];
    #pragma unroll
    for (int i = 0; i < 32; ++i) a = fmaf(myrow[40 + i], prm.cWo[r * 32 + i], a);
    if (gp < prm.n) prm.out[gp * 4 + r] = a;
  }
}

extern "C" void kernel_launch(void* const* d_in, const int* in_sizes, int n_in,
                              void* d_out, int out_size, void* d_ws, size_t ws_size,
                              hipStream_t stream) {
  Params prm;
  prm.p     = (const float*)d_in[0];
  prm.bound = (const float*)d_in[1];
  for (int i = 0; i < 12; ++i) prm.pl[i] = (const float*)d_in[2 + i];
  prm.W0  = (const float*)d_in[14]; prm.b0  = (const float*)d_in[15];
  prm.W1  = (const float*)d_in[16]; prm.b1  = (const float*)d_in[17];
  prm.Wo  = (const float*)d_in[18]; prm.bo  = (const float*)d_in[19];
  prm.cW0 = (const float*)d_in[20]; prm.cb0 = (const float*)d_in[21];
  prm.cW1 = (const float*)d_in[22]; prm.cb1 = (const float*)d_in[23];
  prm.cWo = (const float*)d_in[24]; prm.cbo = (const float*)d_in[25];
  prm.out = (float*)d_out;
  prm.n   = in_sizes[0] / 3;
  const int grid = (prm.n + BLOCK - 1) / BLOCK;
  decomp_decoders_kernel<<<grid, BLOCK, 0, stream>>>(prm);
}